// GaussianAdaptiveAttention_50525995270845
// MI455X (gfx1250) — compile-verified
//
#include <hip/hip_runtime.h>
#include <math.h>

// GaussianAdaptiveAttention for MI455X (gfx1250).
// Bandwidth-bound: 268 MB HBM traffic @ 23.3 TB/s ~ 11.5 us floor.
// Strategy: TDM (tensor_load_to_lds) stages each 8KB row into LDS once;
// all softmax passes run from registers/LDS. wave32 shuffle reductions.

#define ROW       2048
#define THREADS   256          // 8 waves of 32
#define NG        4
#define EPSF      1e-8f

typedef __attribute__((ext_vector_type(4))) unsigned int v4u;
typedef __attribute__((ext_vector_type(4))) int          v4i;
typedef __attribute__((ext_vector_type(8))) int          v8i;

__device__ __forceinline__ float wave_sum32(float v) {
#pragma unroll
    for (int off = 16; off > 0; off >>= 1) v += __shfl_xor(v, off, 32);
    return v;
}
__device__ __forceinline__ float wave_max32(float v) {
#pragma unroll
    for (int off = 16; off > 0; off >>= 1) v = fmaxf(v, __shfl_xor(v, off, 32));
    return v;
}

__global__ __launch_bounds__(THREADS)
void gaa_rowsoftmax_kernel(const float* __restrict__ x,
                           const float* __restrict__ mean_offsets,
                           const float* __restrict__ c,
                           float* __restrict__ out) {
    __shared__ float lds_x[ROW];     // 8 KB row staging (TDM destination)
    __shared__ float red[16];        // cross-wave reduction scratch

    const unsigned t    = threadIdx.x;
    const unsigned lane = t & 31u;
    const unsigned wave = t >> 5;
    const size_t   row  = blockIdx.x;
    const size_t   base = row * (size_t)ROW;

    // ---- TDM: async DMA of one 2048-f32 row (8 KB) global -> LDS ----------
    // Issued once by wave 0 (TDM ignores EXEC; descriptor lives in SGPRs).
    if (t < 32u) {
        unsigned long long ga = (unsigned long long)(uintptr_t)(x + base);
        unsigned lds_off = (unsigned)(uintptr_t)(&lds_x[0]); // flat LDS addr low 32 = LDS byte offset

        v4u g0;
        g0.x = 0x1u;                                   // count=1 (valid user D#)
        g0.y = lds_off;                                // lds_addr (bytes)
        g0.z = (unsigned)(ga & 0xFFFFFFFFull);         // global_addr[31:0]
        g0.w = (unsigned)((ga >> 32) & 0x01FFFFFFull)  // global_addr[56:32]
             | (2u << 30);                             // type = 2 ("image")

        // Group1: workgroup_mask=0, data_size=2 (4B), no barrier/iterate/pad,
        // tensor_dim0=2048, tensor_dim1=1, tile_dim0=2048, tile_dim1=1,
        // tensor_dim0_stride=2048.
        v8i g1 = { (int)0x00020000, (int)0x08000000,   // mask/dsize | dim0.lo
                   (int)0x00010000, (int)0x08000000,   // dim1=1 | tile_dim0=2048
                   (int)0x00000001, (int)0x00000800,   // tile_dim1=1 | stride0=2048
                   0, 0 };
        v4i gz4  = {0, 0, 0, 0};                       // groups 2/3 unused (<=2D)
        v8i gz8  = {0, 0, 0, 0, 0, 0, 0, 0};           // extra group (6-arg form)

        __builtin_amdgcn_tensor_load_to_lds(g0, g1, gz4, gz4, gz8, 0);
        __builtin_amdgcn_s_wait_tensorcnt(0);          // row resident in LDS
    }
    __syncthreads();

    // ---- Pass A: registers <- LDS, row mean / unbiased var -----------------
    const unsigned j0 = t * 4u;            // elems [4t, 4t+3]
    const unsigned j1 = 1024u + t * 4u;    // elems [1024+4t, ...]
    float4 va = *(const float4*)&lds_x[j0];
    float4 vb = *(const float4*)&lds_x[j1];
    float xv[8] = { va.x, va.y, va.z, va.w, vb.x, vb.y, vb.z, vb.w };

    float s = 0.f, ss = 0.f;
#pragma unroll
    for (int k = 0; k < 8; ++k) { s += xv[k]; ss = fmaf(xv[k], xv[k], ss); }
    s  = wave_sum32(s);
    ss = wave_sum32(ss);
    if (lane == 0u) { red[wave] = s; red[8u + wave] = ss; }
    __syncthreads();
    float sum = 0.f, sumsq = 0.f;
#pragma unroll
    for (int w = 0; w < 8; ++w) { sum += red[w]; sumsq += red[8 + w]; }

    const float mean = sum * (1.0f / ROW);
    const float var  = (sumsq - sum * mean) * (1.0f / (ROW - 1));  // ddof=1
    const float rstd = 1.0f / sqrtf(var + EPSF);

    // ---- Gaussian params (tiny uniform loads; L2-resident) -----------------
    float moff[NG], coef[NG];
#pragma unroll
    for (int i = 0; i < NG; ++i) {
        float cc = fminf(fmaxf(c[i], 0.01f), 10.0f);
        float c2 = fmaf(cc, cc, EPSF);
        coef[i] = -0.5f / c2;          // -(1/(2 c^2)); softmax-invariant const dropped
        moff[i] = mean_offsets[i];
    }

    // ---- Pass B: log-mixture per element, row max --------------------------
    float lm[8];
    float lmax = -3.0e38f;
#pragma unroll
    for (int k = 0; k < 8; ++k) {
        const float xc = xv[k] - mean;
        float acc = 0.f;
#pragma unroll
        for (int i = 0; i < NG; ++i) {
            float y = (xc - moff[i]) * rstd;
            y = fminf(fmaxf(y, -10.0f), 10.0f);
            acc = fmaf(y * y, coef[i], acc);
        }
        lm[k] = acc;
        lmax = fmaxf(lmax, acc);
    }
    lmax = wave_max32(lmax);
    __syncthreads();                       // retire pass-A reads of red[]
    if (lane == 0u) red[wave] = lmax;
    __syncthreads();
    float gmax = red[0];
#pragma unroll
    for (int w = 1; w < 8; ++w) gmax = fmaxf(gmax, red[w]);

    // ---- Pass C: exp + row sum --------------------------------------------
    float es = 0.f;
#pragma unroll
    for (int k = 0; k < 8; ++k) {
        float e = __expf(lm[k] - gmax);    // v_exp_f32 fast path
        lm[k] = e;
        es += e;
    }
    es = wave_sum32(es);
    __syncthreads();
    if (lane == 0u) red[wave] = es;
    __syncthreads();
    float tot = 0.f;
#pragma unroll
    for (int w = 0; w < 8; ++w) tot += red[w];
    const float inv = 1.0f / tot;

    // ---- Pass D: out = x * softmax, coalesced b128 stores ------------------
    float4 oa, ob;
    oa.x = xv[0] * lm[0] * inv;  oa.y = xv[1] * lm[1] * inv;
    oa.z = xv[2] * lm[2] * inv;  oa.w = xv[3] * lm[3] * inv;
    ob.x = xv[4] * lm[4] * inv;  ob.y = xv[5] * lm[5] * inv;
    ob.z = xv[6] * lm[6] * inv;  ob.w = xv[7] * lm[7] * inv;
    *(float4*)&out[base + j0] = oa;
    *(float4*)&out[base + j1] = ob;
}

extern "C" void kernel_launch(void* const* d_in, const int* in_sizes, int n_in,
                              void* d_out, int out_size, void* d_ws, size_t ws_size,
                              hipStream_t stream) {
    const float* x  = (const float*)d_in[0];
    const float* mo = (const float*)d_in[1];
    const float* cc = (const float*)d_in[2];
    float* out      = (float*)d_out;

    const int total = in_sizes[0];        // 8*2048*2048
    const int nrows = total / ROW;        // 16384 rows

    gaa_rowsoftmax_kernel<<<dim3(nrows), dim3(THREADS), 0, stream>>>(x, mo, cc, out);
}